// PolygonCanopyLoss_60370060312890
// MI455X (gfx1250) — compile-verified
//
#include <hip/hip_runtime.h>
#include <hip/hip_bf16.h>
#include <math.h>

// ---------------------------------------------------------------------------
// PolygonCanopyLoss = 0.5 * dice(sigmoid(pred), target)
//                   + 0.5 * chamfer(pred_points, target_points)
//
// Chamfer pairwise d2 computed via V_WMMA_F32_16X16X4_F32:
//   d2[i][j] = |p_i|^2 - 2 p_i.q_j + |q_j|^2
//   A row i  = (px_i, py_i, 1, 0)            (16x4, f32, 2 VGPRs)
//   B col j  = (-2qx_j, -2qy_j, |q_j|^2, 0)  (4x16, f32, 2 VGPRs)
//   C[i][j]  = |p_i|^2                       (16x16 f32 accumulator, 8 VGPRs)
// ---------------------------------------------------------------------------

typedef float v2f __attribute__((ext_vector_type(2)));
typedef float v8f __attribute__((ext_vector_type(8)));

#define SMOOTH 1e-6f
#define DICE_NB 512

// -------------------- dice partial reduction --------------------
__global__ void dice_partial_kernel(const float* __restrict__ pred,
                                    const float* __restrict__ tgt,
                                    int n, float* __restrict__ partial) {
  __shared__ float sI[256], sP[256], sT[256];
  const int tid = threadIdx.x;
  float aI = 0.f, aP = 0.f, aT = 0.f;
  for (int i = blockIdx.x * blockDim.x + tid; i < n; i += gridDim.x * blockDim.x) {
    const float x = pred[i];
    const float p = 1.0f / (1.0f + __expf(-x));   // sigmoid
    const float t = tgt[i];
    aI += p * t;
    aP += p;
    aT += t;
  }
  sI[tid] = aI; sP[tid] = aP; sT[tid] = aT;
  __syncthreads();
  for (int s = 128; s > 0; s >>= 1) {
    if (tid < s) {
      sI[tid] += sI[tid + s];
      sP[tid] += sP[tid + s];
      sT[tid] += sT[tid + s];
    }
    __syncthreads();
  }
  if (tid == 0) {
    partial[3 * blockIdx.x + 0] = sI[0];
    partial[3 * blockIdx.x + 1] = sP[0];
    partial[3 * blockIdx.x + 2] = sT[0];
  }
}

// -------------------- chamfer directed min via WMMA --------------------
// For each of the n "P" points, computes min_j d2(P_i, Q_j) into outmin[i].
// One wave32 owns a 16-row strip and sweeps all m columns in 16-wide tiles.
__global__ void chamfer_min_kernel(const float* __restrict__ P, int n,
                                   const float* __restrict__ Q, int m,
                                   float* __restrict__ outmin) {
  const int lane  = threadIdx.x & 31;
  const int wave  = blockIdx.x * (blockDim.x >> 5) + (threadIdx.x >> 5);
  const int row0  = wave << 4;              // 16 rows per wave
  if (row0 >= n) return;                    // wave-uniform exit

  const bool lo = (lane < 16);
  const int  pi = row0 + (lane & 15);
  const float px = P[2 * pi];
  const float py = P[2 * pi + 1];

  // A matrix 16x4 f32: lanes 0-15 -> K0,K1 = (px,py); lanes 16-31 -> K2,K3 = (1,0)
  v2f a;
  a.x = lo ? px : 1.0f;
  a.y = lo ? py : 0.0f;

  // C matrix 16x16 f32: C[i][j] = |p_i|^2 ; VGPR r holds row M = r + 8*(lane/16)
  const float pn2 = px * px + py * py;      // valid rows live on lanes 0..15
  v8f cmat;
#pragma unroll
  for (int r = 0; r < 8; ++r)
    cmat[r] = __shfl(pn2, lo ? r : (r + 8), 32);

  v8f acc;
#pragma unroll
  for (int r = 0; r < 8; ++r) acc[r] = 3.402823466e+38f;

  for (int t = 0; t < m; t += 16) {
    const int qi = t + (lane & 15);
    const float qx = Q[2 * qi];
    const float qy = Q[2 * qi + 1];
    // B matrix 4x16 f32: lanes 0-15 -> K0,K1 = (-2qx,-2qy); lanes 16-31 -> K2,K3 = (|q|^2, 0)
    v2f b;
    b.x = lo ? (-2.0f * qx) : (qx * qx + qy * qy);
    b.y = lo ? (-2.0f * qy) : 0.0f;

    // D = A x B + C  -> full 16x16 tile of squared distances
    v8f d = __builtin_amdgcn_wmma_f32_16x16x4_f32(
        /*neg_a=*/false, a, /*neg_b=*/false, b,
        /*c_mod=*/(short)0, cmat, /*reuse_a=*/false, /*reuse_b=*/false);

#pragma unroll
    for (int r = 0; r < 8; ++r) acc[r] = fminf(acc[r], d[r]);
  }

  // Row-min: reduce across the 16 lanes holding the same row (butterfly within half)
#pragma unroll
  for (int off = 1; off <= 8; off <<= 1) {
#pragma unroll
    for (int r = 0; r < 8; ++r)
      acc[r] = fminf(acc[r], __shfl_xor(acc[r], off, 32));
  }

  // lane 0 holds rows row0..row0+7, lane 16 holds rows row0+8..row0+15
  if ((lane & 15) == 0) {
    const int base = row0 + ((lane >> 4) << 3);
#pragma unroll
    for (int r = 0; r < 8; ++r) outmin[base + r] = acc[r];
  }
}

// -------------------- final scalar combine --------------------
__global__ void finalize_kernel(const float* __restrict__ partial, int nb,
                                const float* __restrict__ rowmin, int n,
                                const float* __restrict__ colmin, int m,
                                float* __restrict__ out) {
  __shared__ float s0[256], s1[256], s2[256], s3[256], s4[256];
  const int tid = threadIdx.x;
  float aI = 0.f, aP = 0.f, aT = 0.f, aR = 0.f, aC = 0.f;
  for (int i = tid; i < nb; i += 256) {
    aI += partial[3 * i + 0];
    aP += partial[3 * i + 1];
    aT += partial[3 * i + 2];
  }
  for (int i = tid; i < n; i += 256) aR += rowmin[i];
  for (int i = tid; i < m; i += 256) aC += colmin[i];
  s0[tid] = aI; s1[tid] = aP; s2[tid] = aT; s3[tid] = aR; s4[tid] = aC;
  __syncthreads();
  for (int s = 128; s > 0; s >>= 1) {
    if (tid < s) {
      s0[tid] += s0[tid + s];
      s1[tid] += s1[tid + s];
      s2[tid] += s2[tid + s];
      s3[tid] += s3[tid + s];
      s4[tid] += s4[tid + s];
    }
    __syncthreads();
  }
  if (tid == 0) {
    const float dice = 1.0f - (2.0f * s0[0] + SMOOTH) / (s1[0] + s2[0] + SMOOTH);
    const float chamfer = s3[0] / (float)n + s4[0] / (float)m;
    out[0] = 0.5f * dice + 0.5f * chamfer;
  }
}

// -------------------- launch --------------------
extern "C" void kernel_launch(void* const* d_in, const int* in_sizes, int n_in,
                              void* d_out, int out_size, void* d_ws, size_t ws_size,
                              hipStream_t stream) {
  const float* pred = (const float*)d_in[0];  // (4,1,1024,1024) logits
  const float* tgt  = (const float*)d_in[1];  // (4,1,1024,1024)
  const float* pp   = (const float*)d_in[2];  // (1,N,2)
  const float* tp   = (const float*)d_in[3];  // (1,M,2)
  float* out = (float*)d_out;

  const int total = in_sizes[0];
  const int N = in_sizes[2] / 2;
  const int M = in_sizes[3] / 2;

  float* ws      = (float*)d_ws;
  float* rowmin  = ws;             // N floats: min over target pts per pred pt
  float* colmin  = ws + N;         // M floats: min over pred pts per target pt
  float* partial = ws + N + M;     // 3*DICE_NB floats

  dice_partial_kernel<<<DICE_NB, 256, 0, stream>>>(pred, tgt, total, partial);

  const int wavesA = (N + 15) / 16;
  const int wavesB = (M + 15) / 16;
  chamfer_min_kernel<<<(wavesA + 7) / 8, 256, 0, stream>>>(pp, N, tp, M, rowmin);
  chamfer_min_kernel<<<(wavesB + 7) / 8, 256, 0, stream>>>(tp, M, pp, N, colmin);

  finalize_kernel<<<1, 256, 0, stream>>>(partial, DICE_NB, rowmin, N, colmin, M, out);
}